// UNet_69277822484491
// MI455X (gfx1250) — compile-verified
//
#include <hip/hip_runtime.h>
#include <hip/hip_bf16.h>

// ---------------------------------------------------------------------------
// GraphUNet (depth 3) for MI455X / gfx1250.
// Heavy op: fused augment+pool  A' = ((A+I)(A+I)) [perm][:,perm], diag=0
// implemented as a gathered fp32 GEMM on V_WMMA_F32_16X16X4_F32.
// ---------------------------------------------------------------------------

typedef float v2f __attribute__((ext_vector_type(2)));
typedef float v8f __attribute__((ext_vector_type(8)));

#define BM 64
#define BN 64
#define BK 32
#define LDP 65   // LDS row pad (64+1) -> conflict-free

// --------------------------------------------------------------------------
// Fused augment + top-k pooling GEMM:
//   C[r,c] = sum_j (A[perm[r],j] + I) * (A[j,perm[c]] + I),  C[r,r] = 0
// A is n x n row-major, C is k x k row-major, perm has k entries.
// --------------------------------------------------------------------------
__global__ __launch_bounds__(256)
void pooled_augment_gemm(const float* __restrict__ A,
                         const int* __restrict__ perm,
                         float* __restrict__ C,
                         int n, int kdim)
{
    __shared__ float As[BK][LDP];   // [k][m]  (K-major for A-frag reads)
    __shared__ float Bs[BK][LDP];   // [k][c]

    const int tid  = threadIdx.x;
    const int lane = tid & 31;
    const int wave = tid >> 5;          // 0..7
    const int wr   = wave & 3;          // wave row: M offset = wr*16
    const int wc   = wave >> 2;         // wave col: N offset = wc*32 (2 tiles)
    const int half = lane >> 4;         // 0: lanes 0-15, 1: lanes 16-31
    const int lm   = lane & 15;

    const int m0 = blockIdx.y * BM;
    const int c0 = blockIdx.x * BN;

    v8f acc0 = {}, acc1 = {};

    const int ksteps = (n + BK - 1) / BK;
    for (int kb = 0; kb < ksteps; ++kb) {
        const int k0 = kb * BK;

        // ---- stage A block: rows gathered through perm, +I folded in ----
        #pragma unroll
        for (int i = 0; i < 8; ++i) {
            int idx = tid + i * 256;          // 0..2047
            int kk  = idx & 31;
            int m   = idx >> 5;
            int gm  = m0 + m;
            int gk  = k0 + kk;
            float v = 0.0f;
            if (gm < kdim && gk < n) {
                int pr = perm[gm];
                v = A[(size_t)pr * n + gk];
                if (pr == gk) v += 1.0f;
            }
            As[kk][m] = v;
        }
        // ---- stage B block: columns gathered through perm, +I folded ----
        #pragma unroll
        for (int i = 0; i < 8; ++i) {
            int idx = tid + i * 256;
            int c   = idx & 63;
            int kk  = idx >> 6;
            int gk  = k0 + kk;
            int gc  = c0 + c;
            float v = 0.0f;
            if (gk < n && gc < kdim) {
                int pc = perm[gc];
                v = A[(size_t)gk * n + pc];
                if (gk == pc) v += 1.0f;
            }
            Bs[kk][c] = v;
        }
        __syncthreads();

        // ---- 8 x (2 WMMA) over the BK=32 slab ----
        #pragma unroll
        for (int kk = 0; kk < BK; kk += 4) {
            v2f a, b0, b1;
            // 16x4 fp32 A frag: lanes 0-15 K=0/1, lanes 16-31 K=2/3
            a.x = As[kk + 2 * half + 0][wr * 16 + lm];
            a.y = As[kk + 2 * half + 1][wr * 16 + lm];
            // 4x16 fp32 B frags (two 16-wide N tiles)
            b0.x = Bs[kk + 2 * half + 0][wc * 32 + lm];
            b0.y = Bs[kk + 2 * half + 1][wc * 32 + lm];
            b1.x = Bs[kk + 2 * half + 0][wc * 32 + 16 + lm];
            b1.y = Bs[kk + 2 * half + 1][wc * 32 + 16 + lm];
            acc0 = __builtin_amdgcn_wmma_f32_16x16x4_f32(
                       false, a, false, b0, (short)0, acc0, false, false);
            acc1 = __builtin_amdgcn_wmma_f32_16x16x4_f32(
                       false, a, false, b1, (short)0, acc1, false, false);
        }
        __syncthreads();
    }

    // ---- store: C/D layout VGPR v -> M = v (lanes 0-15) / v+8 (16-31) ----
    #pragma unroll
    for (int v = 0; v < 8; ++v) {
        int r = m0 + wr * 16 + v + half * 8;
        int c = c0 + wc * 32 + lm;
        if (r < kdim && c < kdim)
            C[(size_t)r * kdim + c] = (r == c) ? 0.0f : acc0[v];
        c = c0 + wc * 32 + 16 + lm;
        if (r < kdim && c < kdim)
            C[(size_t)r * kdim + c] = (r == c) ? 0.0f : acc1[v];
    }
}

// --------------------------------------------------------------------------
// Small helper kernels
// --------------------------------------------------------------------------
__global__ void fill_zero_f32(float* p, size_t n) {
    size_t i = (size_t)blockIdx.x * blockDim.x + threadIdx.x;
    if (i < n) p[i] = 0.0f;
}

__global__ void build_adj(const int* __restrict__ src,
                          const int* __restrict__ dst,
                          float* __restrict__ A, int n, int E)
{
    int e = blockIdx.x * blockDim.x + threadIdx.x;
    if (e >= E) return;
    int s = src[e], d = dst[e];
    if (s != d) atomicAdd(&A[(size_t)s * n + d], 1.0f);
}

// xw[i,c] = sum_f x[i,f] * W[f,c]
__global__ void mm_small(const float* __restrict__ x,
                         const float* __restrict__ W,
                         float* __restrict__ xw,
                         int n, int Fin, int Fout)
{
    int gid = blockIdx.x * blockDim.x + threadIdx.x;
    if (gid >= n * Fout) return;
    int i = gid / Fout, c = gid % Fout;
    float acc = 0.0f;
    for (int f = 0; f < Fin; ++f)
        acc += x[(size_t)i * Fin + f] * W[(size_t)f * Fout + c];
    xw[gid] = acc;
}

// dinv[i] = rsqrt(2 + colsum_i(A))   (A has zero diagonal; Ah = A + 2I)
__global__ void colsum_dinv(const float* __restrict__ A,
                            float* __restrict__ dinv, int n)
{
    int i = blockIdx.x * blockDim.x + threadIdx.x;
    if (i >= n) return;
    float s = 2.0f;
    for (int j = 0; j < n; ++j) s += A[(size_t)j * n + i];
    dinv[i] = rsqrtf(s);
}

__global__ void scale_rows(const float* __restrict__ xw,
                           const float* __restrict__ dinv,
                           float* __restrict__ z, int n, int F)
{
    int gid = blockIdx.x * blockDim.x + threadIdx.x;
    if (gid >= n * F) return;
    z[gid] = dinv[gid / F] * xw[gid];
}

// y[i,c] = dinv[i]*( sum_j A[j,i]*z[j,c] + 2*dinv[i]*xw[i,c] ) + b[c]
__global__ void gcn_agg(const float* __restrict__ A,
                        const float* __restrict__ z,
                        const float* __restrict__ xw,
                        const float* __restrict__ dinv,
                        const float* __restrict__ b,
                        float* __restrict__ y,
                        int n, int F, int relu)
{
    int gid = blockIdx.x * blockDim.x + threadIdx.x;
    if (gid >= n * F) return;
    int i = gid / F, c = gid % F;
    float acc = 0.0f;
    for (int j = 0; j < n; ++j)
        acc += A[(size_t)j * n + i] * z[(size_t)j * F + c];
    float di = dinv[i];
    float val = di * (acc + 2.0f * di * xw[gid]) + b[c];
    if (relu) val = fmaxf(val, 0.0f);
    y[gid] = val;
}

// score[i] = tanh( (x[i,:] . pw) / ||pw|| )
__global__ void pool_score(const float* __restrict__ x,
                           const float* __restrict__ pw,
                           float* __restrict__ score, int n, int H)
{
    int i = blockIdx.x * blockDim.x + threadIdx.x;
    if (i >= n) return;
    float nrm = 0.0f, dot = 0.0f;
    for (int f = 0; f < H; ++f) nrm += pw[f] * pw[f];
    nrm = sqrtf(nrm);
    for (int f = 0; f < H; ++f) dot += x[(size_t)i * H + f] * pw[f];
    score[i] = tanhf(dot / nrm);
}

// rank-based top-k: descending, ties broken toward lower index (== lax.top_k)
__global__ void topk_rank(const float* __restrict__ s,
                          int* __restrict__ perm, int n, int k)
{
    int i = blockIdx.x * blockDim.x + threadIdx.x;
    if (i >= n) return;
    float si = s[i];
    int rank = 0;
    for (int j = 0; j < n; ++j) {
        float sj = s[j];
        rank += (sj > si) || (sj == si && j < i);
    }
    if (rank < k) perm[rank] = i;
}

// xn[r,c] = x[perm[r],c] * score[perm[r]]
__global__ void gather_x(const float* __restrict__ x,
                         const float* __restrict__ score,
                         const int* __restrict__ perm,
                         float* __restrict__ xn, int k, int H)
{
    int gid = blockIdx.x * blockDim.x + threadIdx.x;
    if (gid >= k * H) return;
    int r = gid / H, c = gid % H;
    int p = perm[r];
    xn[gid] = x[(size_t)p * H + c] * score[p];
}

__global__ void copy_f32(const float* __restrict__ a, float* __restrict__ b,
                         size_t n)
{
    size_t i = (size_t)blockIdx.x * blockDim.x + threadIdx.x;
    if (i < n) b[i] = a[i];
}

// xsum[perm[r],c] += xcur[r,c]   (perm entries unique -> no atomics)
__global__ void unpool_add(const float* __restrict__ xcur,
                           const int* __restrict__ perm,
                           float* __restrict__ xsum, int k, int H)
{
    int gid = blockIdx.x * blockDim.x + threadIdx.x;
    if (gid >= k * H) return;
    int r = gid / H, c = gid % H;
    xsum[(size_t)perm[r] * H + c] += xcur[gid];
}

// in-place row-wise log_softmax, one thread per row (C is tiny: 13)
__global__ void log_softmax_rows(float* __restrict__ y, int n, int C)
{
    int i = blockIdx.x * blockDim.x + threadIdx.x;
    if (i >= n) return;
    float* row = y + (size_t)i * C;
    float mx = row[0];
    for (int c = 1; c < C; ++c) mx = fmaxf(mx, row[c]);
    float s = 0.0f;
    for (int c = 0; c < C; ++c) s += expf(row[c] - mx);
    float lse = mx + logf(s);
    for (int c = 0; c < C; ++c) row[c] -= lse;
}

// --------------------------------------------------------------------------
// Host-side orchestration
// --------------------------------------------------------------------------
static inline int cdiv(long a, long b) { return (int)((a + b - 1) / b); }

static void run_gcn(const float* A, int n, const float* x, int Fin,
                    const float* W, const float* b, int Fout, int relu,
                    float* y, float* xw, float* z, float* dinv,
                    hipStream_t s)
{
    int nf = n * Fout;
    mm_small<<<cdiv(nf, 256), 256, 0, s>>>(x, W, xw, n, Fin, Fout);
    colsum_dinv<<<cdiv(n, 256), 256, 0, s>>>(A, dinv, n);
    scale_rows<<<cdiv(nf, 256), 256, 0, s>>>(xw, dinv, z, n, Fout);
    gcn_agg<<<cdiv(nf, 256), 256, 0, s>>>(A, z, xw, dinv, b, y, n, Fout, relu);
}

extern "C" void kernel_launch(void* const* d_in, const int* in_sizes, int n_in,
                              void* d_out, int out_size, void* d_ws, size_t ws_size,
                              hipStream_t stream)
{
    const float* pos = (const float*)d_in[0];
    const int*   ei  = (const int*)d_in[1];
    const float* Wd[4] = {(const float*)d_in[2], (const float*)d_in[4],
                          (const float*)d_in[6], (const float*)d_in[8]};
    const float* bd[4] = {(const float*)d_in[3], (const float*)d_in[5],
                          (const float*)d_in[7], (const float*)d_in[9]};
    const float* pw[3] = {(const float*)d_in[10], (const float*)d_in[11],
                          (const float*)d_in[12]};
    const float* Wu[3] = {(const float*)d_in[13], (const float*)d_in[15],
                          (const float*)d_in[17]};
    const float* bu[3] = {(const float*)d_in[14], (const float*)d_in[16],
                          (const float*)d_in[18]};

    const int N = in_sizes[0] / 3;     // 4096
    const int E = in_sizes[1] / 2;     // 131072
    const int H = in_sizes[3];         // 32
    const int C = in_sizes[18];        // 13
    const int ks[3] = {2000, 1000, 500};         // reference KS
    const int dims[4] = {N, ks[0], ks[1], ks[2]};

    // ---- bump allocator over d_ws ----
    char* p = (char*)d_ws;
    auto alloc = [&](size_t bytes) -> void* {
        void* r = (void*)p;
        p += (bytes + 255) & ~(size_t)255;
        return r;
    };
    float* A[4];
    A[0] = (float*)alloc((size_t)N * N * 4);
    A[1] = (float*)alloc((size_t)ks[0] * ks[0] * 4);
    A[2] = (float*)alloc((size_t)ks[1] * ks[1] * 4);
    A[3] = (float*)alloc((size_t)ks[2] * ks[2] * 4);
    float* X[4];
    for (int l = 0; l < 4; ++l) X[l] = (float*)alloc((size_t)dims[l] * H * 4);
    float* xg   = (float*)alloc((size_t)ks[0] * H * 4);   // pooled-gather x
    float* xsum = (float*)alloc((size_t)N * H * 4);       // skip + unpool sum
    float* U1   = (float*)alloc((size_t)ks[1] * H * 4);   // up outputs
    float* U2   = (float*)alloc((size_t)ks[0] * H * 4);
    float* xw   = (float*)alloc((size_t)N * H * 4);
    float* zb   = (float*)alloc((size_t)N * H * 4);
    float* dinv = (float*)alloc((size_t)N * 4);
    float* scr  = (float*)alloc((size_t)N * 4);
    int*   perm[3];
    for (int l = 0; l < 3; ++l) perm[l] = (int*)alloc((size_t)ks[l] * 4);

    // ---- build dense adjacency (self loops dropped) ----
    size_t nn = (size_t)N * N;
    fill_zero_f32<<<cdiv((long)nn, 256), 256, 0, stream>>>(A[0], nn);
    build_adj<<<cdiv(E, 256), 256, 0, stream>>>(ei, ei + E, A[0], N, E);

    // ---- down path ----
    run_gcn(A[0], N, pos, 3, Wd[0], bd[0], H, 1, X[0], xw, zb, dinv, stream);

    for (int i = 0; i < 3; ++i) {
        int n = dims[i], k = ks[i];
        // scores + perm from x only (augment does not affect them)
        pool_score<<<cdiv(n, 256), 256, 0, stream>>>(X[i], pw[i], scr, n, H);
        topk_rank<<<cdiv(n, 256), 256, 0, stream>>>(scr, perm[i], n, k);
        // fused augment + pool:  A[i+1] = ((A[i]+I)^2)[perm][:,perm], diag=0
        dim3 g(cdiv(k, BN), cdiv(k, BM));
        pooled_augment_gemm<<<g, 256, 0, stream>>>(A[i], perm[i], A[i + 1], n, k);
        // pooled features
        gather_x<<<cdiv(k * H, 256), 256, 0, stream>>>(X[i], scr, perm[i], xg, k, H);
        run_gcn(A[i + 1], k, xg, H, Wd[i + 1], bd[i + 1], H, 1, X[i + 1],
                xw, zb, dinv, stream);
    }

    // ---- up path ----
    const float* xcur = X[3];
    for (int i = 0; i < 3; ++i) {
        int j = 2 - i;                 // skip level
        int nj = dims[j], kj = ks[j];
        copy_f32<<<cdiv((long)nj * H, 256), 256, 0, stream>>>(X[j], xsum,
                                                              (size_t)nj * H);
        unpool_add<<<cdiv(kj * H, 256), 256, 0, stream>>>(xcur, perm[j], xsum,
                                                          kj, H);
        if (i < 2) {
            float* out = (i == 0) ? U1 : U2;
            run_gcn(A[j], nj, xsum, H, Wu[i], bu[i], H, 1, out,
                    xw, zb, dinv, stream);
            xcur = out;
        } else {
            run_gcn(A[0], N, xsum, H, Wu[2], bu[2], C, 0, (float*)d_out,
                    xw, zb, dinv, stream);
        }
    }

    log_softmax_rows<<<cdiv(N, 256), 256, 0, stream>>>((float*)d_out, N, C);
}